// RegDGCNN_48086453846656
// MI455X (gfx1250) — compile-verified
//
#include <hip/hip_runtime.h>

#define EPS   1e-5f
#define KNN   20
#define NPTS  4096
#define BATCH 2
#define RTOT  (BATCH * NPTS)

typedef __attribute__((ext_vector_type(2))) float v2f;
typedef __attribute__((ext_vector_type(8))) float v8f;

// gfx1250 async global->LDS path (guarded; falls back to plain LDS staging).
#if defined(__has_builtin)
#if __has_builtin(__builtin_amdgcn_global_load_async_to_lds_b128) && \
    __has_builtin(__builtin_amdgcn_s_wait_asynccnt)
#define HAVE_ASYNC_LDS 1
#endif
#endif
#ifndef HAVE_ASYNC_LDS
#define HAVE_ASYNC_LDS 0
#endif

// Exact parameter types for the async-LDS builtin (from clang's diagnostic):
// param0 = 16-byte int vector in AS1 (global), param1 = same in AS3 (LDS).
typedef int v4i_raw __attribute__((vector_size(4 * sizeof(int))));
typedef __attribute__((address_space(1))) v4i_raw* gvec_p;
typedef __attribute__((address_space(3))) v4i_raw* lvec_p;

// D = A(16x4 f32) * B(4x16 f32) + C(16x16 f32), full fp32 matrix path on CDNA5.
__device__ __forceinline__ v8f wmma_f32_4(v2f a, v2f b, v8f c) {
  return __builtin_amdgcn_wmma_f32_16x16x4_f32(false, a, false, b, (short)0, c,
                                               false, false);
}

template <int N>
__device__ __forceinline__ void wait_async() {
#if HAVE_ASYNC_LDS
  __builtin_amdgcn_s_wait_asynccnt(N);
#endif
}

// Stage one 16-row x C-col candidate tile into LDS (row stride C+4 floats to
// spread the per-lane A-fragment ds_load_b64's across banks).
template <int C>
__device__ __forceinline__ void stage_tile(const float* __restrict__ Xb,
                                           int ldx, int m0,
                                           float* __restrict__ dst) {
  constexpr int CHT = 16 * (C / 4);  // float4 chunks in one tile
  for (int ch = threadIdx.x; ch < CHT; ch += 256) {
    const int    row = ch / (C / 4);
    const int    c4  = ch - row * (C / 4);
    const float* gp  = Xb + (size_t)(m0 + row) * ldx + c4 * 4;
    float*       lp  = dst + row * (C + 4) + c4 * 4;
#if HAVE_ASYNC_LDS
    __builtin_amdgcn_global_load_async_to_lds_b128((gvec_p)gp, (lvec_p)lp, 0, 0);
#else
    *(float4*)lp = *(const float4*)gp;
#endif
  }
}

// ---------------------------------------------------------------------------
// KNN: 8 waves / workgroup; each wave owns 16 query rows. Candidate tiles are
// double-buffered in LDS (async prefetch one tile ahead) and shared by all 8
// waves; pairwise inner products via fp32 WMMA; per-lane register top-20;
// lane halves merged through LDS. neg_dist = 2*inner - sq[q] - sq[m].
// ---------------------------------------------------------------------------
template <int C>
__global__ void __launch_bounds__(256)
knn_kernel(const float* __restrict__ X, int ldx, const float* __restrict__ sq,
           int* __restrict__ knn_out) {
  constexpr int LSTR = C + 4;
  constexpr int CHT  = 16 * (C / 4);
  constexpr int NCH  = (CHT + 255) / 256;  // async instrs per wave per stage
  constexpr int NT   = NPTS / 16;

  __shared__ float tile[2][16 * LSTR];
  __shared__ float sv[256][KNN];
  __shared__ int   si[256][KNN];

  const int tid  = threadIdx.x;
  const int wv   = tid >> 5;
  const int lane = tid & 31;
  const int half = lane >> 4;  // 0: K=0,1 / M rows 0..7 ; 1: K=2,3 / rows 8..15
  const int l16  = lane & 15;
  const int koff = half * 2;
  const int b    = blockIdx.y;
  const int q0   = (blockIdx.x * 8 + wv) * 16;

  const float* Xb  = X  + (size_t)b * NPTS * ldx;
  const float* sqb = sq + (size_t)b * NPTS;

  // B-matrix (queries, 4x16 per K-step) fragments, kept in registers.
  v2f bq[C / 4];
#pragma unroll
  for (int t = 0; t < C / 4; ++t)
    bq[t] = *(const v2f*)(Xb + (size_t)(q0 + l16) * ldx + 4 * t + koff);
  const float sqq = sqb[q0 + l16];

  float bestv[KNN];
  int   besti[KNN];
#pragma unroll
  for (int j = 0; j < KNN; ++j) {
    bestv[j] = -3.0e38f;
    besti[j] = 0;
  }

  stage_tile<C>(Xb, ldx, 0, tile[0]);
  for (int it = 0; it < NT; ++it) {
    const int cur = it & 1;
    if (it + 1 < NT) {
      stage_tile<C>(Xb, ldx, (it + 1) * 16, tile[cur ^ 1]);
      wait_async<NCH>();  // in-order: tile `it` is complete
    } else {
      wait_async<0>();
    }
    __syncthreads();

    v8f acc = {};
#pragma unroll
    for (int t = 0; t < C / 4; ++t) {
      v2f a = *(const v2f*)&tile[cur][l16 * LSTR + 4 * t + koff];
      acc   = wmma_f32_4(a, bq[t], acc);
    }
    const int m0 = it * 16;
    // This lane sees candidates m0 + j + 8*half for query (q0 + l16).
#pragma unroll
    for (int j = 0; j < 8; ++j) {
      const int   cand = m0 + j + 8 * half;
      const float nd   = 2.0f * acc[j] - sqq - sqb[cand];
      if (nd > bestv[KNN - 1]) {
        float cv = nd;
        int   ci = cand;
#pragma unroll
        for (int t = 0; t < KNN; ++t) {
          const bool  gt = (cv > bestv[t]);
          const float tv = bestv[t];
          const int   ti = besti[t];
          bestv[t] = gt ? cv : tv;
          besti[t] = gt ? ci : ti;
          cv       = gt ? tv : cv;
          ci       = gt ? ti : ci;
        }
      }
    }
    __syncthreads();
  }

  // Merge the two per-lane sorted lists of each query via LDS.
#pragma unroll
  for (int j = 0; j < KNN; ++j) {
    sv[tid][j] = bestv[j];
    si[tid][j] = besti[j];
  }
  __syncthreads();
  if (lane < 16) {
    int  i0  = 0, i1 = 0;
    int* out = knn_out + ((size_t)b * NPTS + q0 + l16) * KNN;
    for (int j = 0; j < KNN; ++j) {
      const float v0 = sv[tid][i0], v1 = sv[tid + 16][i1];
      const int   d0 = si[tid][i0], d1 = si[tid + 16][i1];
      const bool  t0 = (v0 > v1) || (v0 == v1 && d0 < d1);
      out[j] = t0 ? d0 : d1;
      if (t0) ++i0; else ++i1;
    }
  }
}

// ---------------------------------------------------------------------------
// fp32 WMMA GEMM, 32 rows x 64 cols per wave (acc[2][4], 8 WMMAs / 6 loads
// per K-step): Out[r, o] = sum_k A[r, k] * W[o, wcol0 + k].
// MODE 0: raw store. MODE 1: + bias, batchnorm, ReLU (head layers).
// ---------------------------------------------------------------------------
template <int MODE>
__global__ void __launch_bounds__(32)
gemm_kernel(const float* __restrict__ A, int lda, const float* __restrict__ W,
            int ldw, int wcol0, float* __restrict__ Out, int ldo, int K,
            const float* __restrict__ bias, const float* __restrict__ g,
            const float* __restrict__ bt, const float* __restrict__ m,
            const float* __restrict__ v) {
  const int lane = threadIdx.x;
  const int half = lane >> 4;
  const int l16  = lane & 15;
  const int koff = half * 2;
  const int r0   = blockIdx.x * 32;
  const int o0   = blockIdx.y * 64;

  const float* ar0 = A + (size_t)(r0 + l16) * lda + koff;
  const float* ar1 = ar0 + (size_t)16 * lda;
  const float* wr[4];
#pragma unroll
  for (int j = 0; j < 4; ++j)
    wr[j] = W + (size_t)(o0 + 16 * j + l16) * ldw + wcol0 + koff;

  v8f acc[2][4] = {};
  for (int t = 0; t < K; t += 4) {
    const v2f a0 = *(const v2f*)(ar0 + t);
    const v2f a1 = *(const v2f*)(ar1 + t);
#pragma unroll
    for (int j = 0; j < 4; ++j) {
      const v2f bb = *(const v2f*)(wr[j] + t);
      acc[0][j]    = wmma_f32_4(a0, bb, acc[0][j]);
      acc[1][j]    = wmma_f32_4(a1, bb, acc[1][j]);
    }
  }

#pragma unroll
  for (int j = 0; j < 4; ++j) {
    const int o     = o0 + 16 * j + l16;
    float     scale = 1.0f, shift = 0.0f, bi = 0.0f;
    if (MODE == 1) {
      bi             = bias[o];
      const float rs = rsqrtf(v[o] + EPS);
      scale          = rs * g[o];
      shift          = bt[o] - m[o] * scale;
    }
#pragma unroll
    for (int i = 0; i < 2; ++i) {
#pragma unroll
      for (int p = 0; p < 8; ++p) {
        const int r = r0 + 16 * i + p + 8 * half;
        float     h = acc[i][j][p];
        if (MODE == 1) {
          h = (h + bi) * scale + shift;
          h = h > 0.0f ? h : 0.0f;
        }
        Out[(size_t)r * ldo + o] = h;
      }
    }
  }
}

// ---------------------------------------------------------------------------
// Edge-conv epilogue: h[k] = P[idx_k] - P[r] + Q[r]; BN; leaky ReLU; max over k.
// Writes into the concat buffer at column offset col0.
// ---------------------------------------------------------------------------
__global__ void gather_max_kernel(const float* __restrict__ P,
                                  const float* __restrict__ Q,
                                  const int* __restrict__ idx, int co,
                                  float* __restrict__ Out, int ldo, int col0,
                                  const float* __restrict__ g,
                                  const float* __restrict__ bt,
                                  const float* __restrict__ m,
                                  const float* __restrict__ v) {
  const int tid = blockIdx.x * blockDim.x + threadIdx.x;
  if (tid >= RTOT * co) return;
  const int r = tid / co;
  const int o = tid - r * co;
  const int b = r / NPTS;

  const float* Pb   = P + (size_t)b * NPTS * co;
  const float  base = Q[(size_t)r * co + o] - P[(size_t)r * co + o];
  const float  rs   = rsqrtf(v[o] + EPS);
  const float  sc   = rs * g[o];
  const float  sh   = bt[o] - m[o] * sc;
  const int*   id   = idx + (size_t)r * KNN;

  float best = -3.0e38f;
#pragma unroll 4
  for (int k = 0; k < KNN; ++k) {
    float h = Pb[(size_t)id[k] * co + o] + base;
    h       = h * sc + sh;
    h       = h > 0.0f ? h : 0.2f * h;
    best    = fmaxf(best, h);
  }
  Out[(size_t)r * ldo + col0 + o] = best;
}

// ---------------------------------------------------------------------------
// Small helper kernels.
// ---------------------------------------------------------------------------
__global__ void pad_x_kernel(const float* __restrict__ x, float* __restrict__ xp) {
  const int r = blockIdx.x * blockDim.x + threadIdx.x;
  if (r >= RTOT) return;
  xp[r * 4 + 0] = x[r * 3 + 0];
  xp[r * 4 + 1] = x[r * 3 + 1];
  xp[r * 4 + 2] = x[r * 3 + 2];
  xp[r * 4 + 3] = 0.0f;
}

__global__ void sumsq_kernel(const float* __restrict__ X, int ldx, int C,
                             float* __restrict__ sq) {
  const int r = blockIdx.x * blockDim.x + threadIdx.x;
  if (r >= RTOT) return;
  const float* p = X + (size_t)r * ldx;
  float        s = 0.0f;
  for (int c = 0; c < C; ++c) s += p[c] * p[c];
  sq[r] = s;
}

// Layer-1 projections (Cin = 3): P = x . w[:, :3]^T, Q = x . w[:, 3:6]^T
__global__ void proj1_kernel(const float* __restrict__ x,
                             const float* __restrict__ w,
                             float* __restrict__ P, float* __restrict__ Q) {
  const int tid = blockIdx.x * blockDim.x + threadIdx.x;
  if (tid >= RTOT * 64) return;
  const int    r  = tid / 64;
  const int    o  = tid - r * 64;
  const float* xr = x + (size_t)r * 3;
  const float* wr = w + (size_t)o * 6;
  P[tid] = xr[0] * wr[0] + xr[1] * wr[1] + xr[2] * wr[2];
  Q[tid] = xr[0] * wr[3] + xr[1] * wr[4] + xr[2] * wr[5];
}

__global__ void final_kernel(const float* __restrict__ h,
                             const float* __restrict__ w,
                             const float* __restrict__ b,
                             float* __restrict__ out) {
  const int r = blockIdx.x * blockDim.x + threadIdx.x;
  if (r >= RTOT) return;
  const float* hr = h + (size_t)r * 128;
  float        s  = b[0];
#pragma unroll 8
  for (int c = 0; c < 128; ++c) s += hr[c] * w[c];
  out[r] = s;
}

// ---------------------------------------------------------------------------
extern "C" void kernel_launch(void* const* d_in, const int* in_sizes, int n_in,
                              void* d_out, int out_size, void* d_ws,
                              size_t ws_size, hipStream_t stream) {
  (void)in_sizes; (void)n_in; (void)out_size; (void)ws_size;

  const float* x = (const float*)d_in[0];
  const float* w1 = (const float*)d_in[1];
  const float* g1 = (const float*)d_in[2];
  const float* bt1 = (const float*)d_in[3];
  const float* m1 = (const float*)d_in[4];
  const float* v1 = (const float*)d_in[5];
  const float* w2 = (const float*)d_in[6];
  const float* g2 = (const float*)d_in[7];
  const float* bt2 = (const float*)d_in[8];
  const float* m2 = (const float*)d_in[9];
  const float* v2 = (const float*)d_in[10];
  const float* w3 = (const float*)d_in[11];
  const float* g3 = (const float*)d_in[12];
  const float* bt3 = (const float*)d_in[13];
  const float* m3 = (const float*)d_in[14];
  const float* v3 = (const float*)d_in[15];
  const float* w4 = (const float*)d_in[16];
  const float* g4 = (const float*)d_in[17];
  const float* bt4 = (const float*)d_in[18];
  const float* m4 = (const float*)d_in[19];
  const float* v4 = (const float*)d_in[20];
  const float* hw1 = (const float*)d_in[21];
  const float* hb1 = (const float*)d_in[22];
  const float* hg1 = (const float*)d_in[23];
  const float* hbt1 = (const float*)d_in[24];
  const float* hm1 = (const float*)d_in[25];
  const float* hv1 = (const float*)d_in[26];
  const float* hw2 = (const float*)d_in[27];
  const float* hb2 = (const float*)d_in[28];
  const float* hg2 = (const float*)d_in[29];
  const float* hbt2 = (const float*)d_in[30];
  const float* hm2 = (const float*)d_in[31];
  const float* hv2 = (const float*)d_in[32];
  const float* hw3 = (const float*)d_in[33];
  const float* hb3 = (const float*)d_in[34];

  // Workspace carve-out (~47 MB total).
  char* wsp = (char*)d_ws;
  auto  carve = [&](size_t bytes) -> void* {
    void* r = (void*)wsp;
    wsp += (bytes + 255) & ~(size_t)255;
    return r;
  };
  float* xpad = (float*)carve((size_t)RTOT * 4 * 4);
  float* sq   = (float*)carve((size_t)RTOT * 4);
  int*   idx  = (int*)carve((size_t)RTOT * KNN * 4);
  float* P    = (float*)carve((size_t)RTOT * 256 * 4);
  float* Q    = (float*)carve((size_t)RTOT * 256 * 4);
  float* xc   = (float*)carve((size_t)RTOT * 512 * 4);
  float* h1   = (float*)carve((size_t)RTOT * 256 * 4);
  float* h2   = (float*)carve((size_t)RTOT * 128 * 4);

  const dim3 knn_grid(NPTS / 128, BATCH);  // 8 waves x 16 queries per WG
  const int  TB = 256;
  const int  gR = (RTOT + TB - 1) / TB;
  const float* nul = nullptr;

  // ---- Layer 1 (Cin=3 -> 64), KNN on zero-padded x (K=4) ----
  pad_x_kernel<<<gR, TB, 0, stream>>>(x, xpad);
  sumsq_kernel<<<gR, TB, 0, stream>>>(xpad, 4, 4, sq);
  knn_kernel<4><<<knn_grid, 256, 0, stream>>>(xpad, 4, sq, idx);
  proj1_kernel<<<(RTOT * 64 + TB - 1) / TB, TB, 0, stream>>>(x, w1, P, Q);
  gather_max_kernel<<<(RTOT * 64 + TB - 1) / TB, TB, 0, stream>>>(
      P, Q, idx, 64, xc, 512, 0, g1, bt1, m1, v1);

  // ---- Layer 2 (x1: xc cols [0,64), Cin=64 -> 64) ----
  sumsq_kernel<<<gR, TB, 0, stream>>>(xc, 512, 64, sq);
  knn_kernel<64><<<knn_grid, 256, 0, stream>>>(xc, 512, sq, idx);
  gemm_kernel<0><<<dim3(RTOT / 32, 1), 32, 0, stream>>>(
      xc, 512, w2, 128, 0, P, 64, 64, nul, nul, nul, nul, nul);
  gemm_kernel<0><<<dim3(RTOT / 32, 1), 32, 0, stream>>>(
      xc, 512, w2, 128, 64, Q, 64, 64, nul, nul, nul, nul, nul);
  gather_max_kernel<<<(RTOT * 64 + TB - 1) / TB, TB, 0, stream>>>(
      P, Q, idx, 64, xc, 512, 64, g2, bt2, m2, v2);

  // ---- Layer 3 (x2: xc cols [64,128), Cin=64 -> 128) ----
  sumsq_kernel<<<gR, TB, 0, stream>>>(xc + 64, 512, 64, sq);
  knn_kernel<64><<<knn_grid, 256, 0, stream>>>(xc + 64, 512, sq, idx);
  gemm_kernel<0><<<dim3(RTOT / 32, 2), 32, 0, stream>>>(
      xc + 64, 512, w3, 128, 0, P, 128, 64, nul, nul, nul, nul, nul);
  gemm_kernel<0><<<dim3(RTOT / 32, 2), 32, 0, stream>>>(
      xc + 64, 512, w3, 128, 64, Q, 128, 64, nul, nul, nul, nul, nul);
  gather_max_kernel<<<(RTOT * 128 + TB - 1) / TB, TB, 0, stream>>>(
      P, Q, idx, 128, xc, 512, 128, g3, bt3, m3, v3);

  // ---- Layer 4 (x3: xc cols [128,256), Cin=128 -> 256) ----
  sumsq_kernel<<<gR, TB, 0, stream>>>(xc + 128, 512, 128, sq);
  knn_kernel<128><<<knn_grid, 256, 0, stream>>>(xc + 128, 512, sq, idx);
  gemm_kernel<0><<<dim3(RTOT / 32, 4), 32, 0, stream>>>(
      xc + 128, 512, w4, 256, 0, P, 256, 128, nul, nul, nul, nul, nul);
  gemm_kernel<0><<<dim3(RTOT / 32, 4), 32, 0, stream>>>(
      xc + 128, 512, w4, 256, 128, Q, 256, 128, nul, nul, nul, nul, nul);
  gather_max_kernel<<<(RTOT * 256 + TB - 1) / TB, TB, 0, stream>>>(
      P, Q, idx, 256, xc, 512, 256, g4, bt4, m4, v4);

  // ---- Head: 512 -> 256 -> 128 -> 1 ----
  gemm_kernel<1><<<dim3(RTOT / 32, 4), 32, 0, stream>>>(
      xc, 512, hw1, 512, 0, h1, 256, 512, hb1, hg1, hbt1, hm1, hv1);
  gemm_kernel<1><<<dim3(RTOT / 32, 2), 32, 0, stream>>>(
      h1, 256, hw2, 256, 0, h2, 128, 256, hb2, hg2, hbt2, hm2, hv2);
  final_kernel<<<gR, TB, 0, stream>>>(h2, hw3, hb3, (float*)d_out);
}